// PatchySANPooling_90271622627425
// MI455X (gfx1250) — compile-verified
//
#include <hip/hip_runtime.h>
#include <math.h>

#define NDIM   256   // feature dim D
#define NGRAPH 512   // number of graphs G
#define TOPK   64    // K
#define CAP    2048  // max segment norms staged in LDS (actual max ~460)

typedef __attribute__((ext_vector_type(2))) float v2f;
typedef __attribute__((ext_vector_type(8))) float v8f;

// ---------------------------------------------------------------------------
// Kernel 1: row L2 norms via WMMA Gram-diagonal trick.
// One wave handles 16 rows. The 16x4 f32 A-tile layout (lane=row,
// lane-half/VGPR = K) equals the B-layout of its own transpose, so we pass the
// same registers as A and B and accumulate diag(X·X^T) over 64 K-chunks.
// ---------------------------------------------------------------------------
__global__ __launch_bounds__(256) void norms_kernel(const float* __restrict__ x,
                                                    float* __restrict__ norms,
                                                    int N)
{
    const int  lane    = threadIdx.x & 31;
    const int  wave    = threadIdx.x >> 5;
    const long rowBase = (long)blockIdx.x * 128 + (long)wave * 16;
    const int  mrow    = lane & 15;
    const int  khalf   = lane >> 4;          // 0 -> cols {4k,4k+1}, 1 -> {4k+2,4k+3}
    long row = rowBase + mrow;
    if (row >= N) row = N - 1;               // clamp: keep EXEC uniform for loads/WMMA
    const float* rp = x + row * NDIM + khalf * 2;

#if __has_builtin(__builtin_amdgcn_wmma_f32_16x16x4_f32)
    v8f acc0 = {0.f,0.f,0.f,0.f,0.f,0.f,0.f,0.f};
    v8f acc1 = {0.f,0.f,0.f,0.f,0.f,0.f,0.f,0.f};
    #pragma unroll 4
    for (int k = 0; k < NDIM; k += 8) {
        float2 f0 = *(const float2*)(rp + k);
        float2 f1 = *(const float2*)(rp + k + 4);
        v2f a0; a0.x = f0.x; a0.y = f0.y;
        v2f a1; a1.x = f1.x; a1.y = f1.y;
        acc0 = __builtin_amdgcn_wmma_f32_16x16x4_f32(false, a0, false, a0,
                                                     (short)0, acc0, false, false);
        acc1 = __builtin_amdgcn_wmma_f32_16x16x4_f32(false, a1, false, a1,
                                                     (short)0, acc1, false, false);
    }
    v8f acc = acc0 + acc1;

    // Diagonal (m,m) of the 16x16 f32 C/D tile:
    //   m in 0..7  -> VGPR m,   lane m
    //   m in 8..15 -> VGPR m-8, lane m+16
    const bool valid = (lane < 8) || (lane >= 24);
    const int  m  = (lane < 8) ? lane : (lane - 16);
    const int  ci = m & 7;
    float d = acc[0];
    if (ci == 1) d = acc[1];
    if (ci == 2) d = acc[2];
    if (ci == 3) d = acc[3];
    if (ci == 4) d = acc[4];
    if (ci == 5) d = acc[5];
    if (ci == 6) d = acc[6];
    if (ci == 7) d = acc[7];
    long orow = rowBase + m;
    if (valid && orow < N) norms[orow] = sqrtf(d);
#else
    // Fallback (also what the host pass parses): lane L and L+16 share row L.
    float s = 0.f;
    #pragma unroll 4
    for (int k = 0; k < NDIM; k += 8) {
        float2 f0 = *(const float2*)(rp + k);
        float2 f1 = *(const float2*)(rp + k + 4);
        s += f0.x*f0.x + f0.y*f0.y + f1.x*f1.x + f1.y*f1.y;
    }
    s += __shfl_down(s, 16, 32);
    long orow = rowBase + mrow;
    if (khalf == 0 && orow < N) norms[orow] = sqrtf(s);
#endif
}

// ---------------------------------------------------------------------------
// Kernel 2: segment starts via binary search (batch is sorted ascending).
// seg[g] = lower_bound(batch, g), seg[NGRAPH] = N.
// ---------------------------------------------------------------------------
__global__ void segstart_kernel(const int* __restrict__ batch,
                                int* __restrict__ seg, int N)
{
    int g = blockIdx.x * blockDim.x + threadIdx.x;
    if (g > NGRAPH) return;
    if (g == NGRAPH) { seg[NGRAPH] = N; return; }
    int lo = 0, hi = N;
    while (lo < hi) {
        int mid = (lo + hi) >> 1;
        if (batch[mid] < g) lo = mid + 1; else hi = mid;
    }
    seg[g] = lo;
}

// ---------------------------------------------------------------------------
// Kernel 3: one block per graph. Rank-count top-K selection (norm desc,
// index asc tiebreak -> ranks are a permutation, so topidx writes are
// conflict-free), then coalesced float4 gather with zero padding.
// ---------------------------------------------------------------------------
__global__ __launch_bounds__(256) void topk_gather_kernel(const float* __restrict__ x,
                                                          const float* __restrict__ norms,
                                                          const int* __restrict__ seg,
                                                          float* __restrict__ out)
{
    __shared__ float snorm[CAP];
    __shared__ int   topidx[TOPK];

    const int g     = blockIdx.x;
    const int start = seg[g];
    const int end   = seg[g + 1];
    const int len   = end - start;
    const int t     = threadIdx.x;
    const int lcap  = (len < CAP) ? len : CAP;

    for (int i = t; i < lcap; i += 256) snorm[i] = norms[start + i];
    __syncthreads();

    for (int j = t; j < len; j += 256) {
        const float nj = (j < CAP) ? snorm[j] : norms[start + j];
        int rank = 0;
        for (int i = 0; i < lcap && rank < TOPK; ++i) {
            const float ni = snorm[i];
            rank += (ni > nj) || ((ni == nj) && (i < j));
        }
        for (int i = lcap; i < len && rank < TOPK; ++i) {
            const float ni = norms[start + i];
            rank += (ni > nj) || ((ni == nj) && (i < j));
        }
        if (rank < TOPK) topidx[rank] = start + j;
    }
    __syncthreads();

    const int nvalid = (len < TOPK) ? len : TOPK;
    float4* og = (float4*)(out + (size_t)g * (TOPK * NDIM));
    // 4 rows per pass, 64 threads (one float4 each) per 256-float row.
    for (int r4 = 0; r4 < TOPK; r4 += 4) {
        const int r = r4 + (t >> 6);
        const int c = t & 63;
        float4 v = make_float4(0.f, 0.f, 0.f, 0.f);
        if (r < nvalid) {
            const float4* src = (const float4*)(x + (size_t)topidx[r] * NDIM);
            v = src[c];
        }
        og[r * (NDIM / 4) + c] = v;
    }
}

// ---------------------------------------------------------------------------
extern "C" void kernel_launch(void* const* d_in, const int* in_sizes, int n_in,
                              void* d_out, int out_size, void* d_ws, size_t ws_size,
                              hipStream_t stream)
{
    const float* x     = (const float*)d_in[0];   // [N, 256] f32
    const int*   batch = (const int*)d_in[1];     // [N] i32, sorted
    float*       out   = (float*)d_out;           // [512, 64*256] f32
    const int N = in_sizes[1];

    // Scratch layout in d_ws: norms[N] f32, then seg[G+1] i32.
    float* norms = (float*)d_ws;
    int*   seg   = (int*)((char*)d_ws + (((size_t)N * sizeof(float) + 255) & ~(size_t)255));

    // 1) row norms: 16 rows per wave, 128 rows per 256-thread block
    norms_kernel<<<(N + 127) / 128, 256, 0, stream>>>(x, norms, N);
    // 2) segment starts: G+1 binary searches
    segstart_kernel<<<(NGRAPH + 1 + 255) / 256, 256, 0, stream>>>(batch, seg, N);
    // 3) per-graph top-K select + gather + pad
    topk_gather_kernel<<<NGRAPH, 256, 0, stream>>>(x, norms, seg, out);
}